// AnchorStores_3573412790449
// MI455X (gfx1250) — compile-verified
//
#include <hip/hip_runtime.h>
#include <hip/hip_bf16.h>

#define B_ 8
#define K_ 256
#define DIM_ 50257
#define KNN_ 8
#define NCLASS_ 4
#define KNN_T_ 0.05f

#define DCHUNKS 32          // DIM split into 32 chunks -> 4096 waves total
#define WAVES_PER_BLOCK 8   // 256-thread blocks (8 wave32s)

#define LN2_ 0.69314718055994530942f

typedef float v2f __attribute__((ext_vector_type(2)));
typedef float v8f __attribute__((ext_vector_type(8)));

// ---------------------------------------------------------------------------
// Kernel 1: ll[b,d] = log(logits[b,d]); zero the per-(b,k) partial-sum slots.
// Inputs are in [1e-3, 1] -> raw v_log_f32 (log2) * ln2, no denorm/inf guards.
// ---------------------------------------------------------------------------
__global__ __launch_bounds__(256) void kl_init_kernel(
    const float* __restrict__ logits, float* __restrict__ ll,
    float* __restrict__ dists) {
  int i = blockIdx.x * blockDim.x + threadIdx.x;
  if (i < B_ * DIM_) ll[i] = LN2_ * __builtin_amdgcn_logf(logits[i]);
  if (i < B_ * K_)   dists[i] = 0.0f;
}

// ---------------------------------------------------------------------------
// Kernel 2: streaming KL reduction (memory-bound: 412 MB read once,
// HBM floor ~= 412MB / 23.3TB/s ~= 17.7us).
//   s1 (TRANS+VALU): per-lane  sum qa*log2(qa)   (scaled by ln2 at the end)
//   S2 (WMMA f32 16x16x4): sum qa*ll as a GEMV:
//     A[m,kk] = ll[d+kk] replicated over m  -> lane L regs = ll[d+off2 .. +1]
//     B[kk,n] = qa[row k0+n, d+kk]          -> lane L regs = qa[row, d+off2 .. +1]
//     => C[m,n] = S2(row n) for every m; lane L's C[0] = S2(row L%16).
//   dists[b,k] += ln2*s1 - 0.5*C[0]  (both lane halves hold row k -> 0.5 each)
// Loop bounds are forced into SGPRs (readfirstlane) -> scalar loop control,
// unrollable, EXEC statically all-ones around every WMMA.
// ---------------------------------------------------------------------------
__global__ __launch_bounds__(256) void kl_wmma_kernel(
    const float* __restrict__ qa, const float* __restrict__ ll,
    float* __restrict__ dists) {
  const int lane = threadIdx.x & 31;
  const int wave = __builtin_amdgcn_readfirstlane(threadIdx.x >> 5); // uniform
  const int tile = blockIdx.x >> 2;                 // 0..127  (b, k-tile)
  const int cgrp = blockIdx.x & 3;
  const int c    = cgrp * WAVES_PER_BLOCK + wave;   // 0..31   DIM chunk id (SGPR)
  const int b    = tile >> 4;
  const int k0   = (tile & 15) << 4;
  const int row  = k0 + (lane & 15);
  const int off2 = (lane >> 4) << 1;                // 0 for lanes 0-15, 2 for 16-31

  const int chunk = (DIM_ + DCHUNKS - 1) / DCHUNKS;
  const int lo    = c * chunk;                      // SGPR
  const int hi    = (lo + chunk < DIM_) ? (lo + chunk) : DIM_;  // SGPR
  const int nfull = (hi - lo) >> 2;                 // SGPR
  const int dfull = lo + (nfull << 2);              // SGPR

  const float* rowp = qa + (size_t)(b * K_ + row) * DIM_;
  const float* llb  = ll + (size_t)b * DIM_;

  v8f   acc = {};       // WMMA C/D accumulator (S2 per row, replicated over m)
  float s1  = 0.0f;     // per-lane sum qa*log2(qa) over this lane's d subset
  float s2t = 0.0f;     // tail-only sum qa*ll

#pragma unroll 4
  for (int d = lo; d < dfull; d += 4) {
    const int dd = d + off2;
    // 412 MB stream, read once: non-temporal. ll: regular temporal (hot in L2).
    v2f q = __builtin_nontemporal_load((const v2f*)(rowp + dd));
    v2f a = *(const v2f*)(llb + dd);
    s1 = __builtin_fmaf(q.x, __builtin_amdgcn_logf(q.x), s1);  // raw v_log_f32
    s1 = __builtin_fmaf(q.y, __builtin_amdgcn_logf(q.y), s1);
    acc = __builtin_amdgcn_wmma_f32_16x16x4_f32(
        /*neg_a=*/false, a, /*neg_b=*/false, q,
        /*c_mod=*/(short)0, acc, /*reuse_a=*/false, /*reuse_b=*/false);
  }

  // Tail (1..3 leftover d's): lanes 0-15 handle their row scalar-wise.
  if (lane < 16) {
    for (int dd = dfull; dd < hi; ++dd) {
      float qv = rowp[dd];
      s1  = __builtin_fmaf(qv, __builtin_amdgcn_logf(qv), s1);
      s2t = __builtin_fmaf(qv, llb[dd], s2t);
    }
  }

  float contrib = LN2_ * s1 - s2t - 0.5f * acc[0];
  atomicAdd(&dists[b * K_ + row], contrib);
}

// ---------------------------------------------------------------------------
// Kernel 3: per-b top-8 (stable: smallest index wins ties, like lax.top_k),
// softmax over the 8, scatter weights into 4 class bins. 8 rows -> trivial.
// ---------------------------------------------------------------------------
__global__ void kl_topk_kernel(const float* __restrict__ dists,
                               const int* __restrict__ labels,
                               float* __restrict__ out) {
  int b = threadIdx.x;
  if (b >= B_) return;
  const float scale = -1.0f / (KNN_T_ * (float)DIM_);  // mean + (-1/T)
  float s[K_];
  for (int k = 0; k < K_; ++k) s[k] = dists[b * K_ + k] * scale;

  float tv[KNN_]; int ti[KNN_];
  const float NEG = -__builtin_huge_valf();
  for (int i = 0; i < KNN_; ++i) {
    float best = NEG; int bi = 0;
    for (int j = 0; j < K_; ++j)
      if (s[j] > best) { best = s[j]; bi = j; }   // strict > => lowest index on tie
    tv[i] = best; ti[i] = bi; s[bi] = NEG;
  }

  float m = tv[0], sum = 0.0f, w[KNN_];
  for (int i = 0; i < KNN_; ++i) { w[i] = __expf(tv[i] - m); sum += w[i]; }
  float p[NCLASS_] = {0.f, 0.f, 0.f, 0.f};
  for (int i = 0; i < KNN_; ++i) p[labels[b * K_ + ti[i]]] += w[i];
  float inv = 1.0f / sum;
  for (int cc = 0; cc < NCLASS_; ++cc) out[b * NCLASS_ + cc] = p[cc] * inv;
}

// ---------------------------------------------------------------------------
extern "C" void kernel_launch(void* const* d_in, const int* in_sizes, int n_in,
                              void* d_out, int out_size, void* d_ws, size_t ws_size,
                              hipStream_t stream) {
  const float* logits = (const float*)d_in[0];   // [B, DIM]
  const float* qa     = (const float*)d_in[1];   // [B, K, DIM]
  const int*   labels = (const int*)d_in[2];     // [B, K]
  float* out = (float*)d_out;                    // [B, NCLASS]

  float* ll    = (float*)d_ws;                   // B*DIM floats
  float* dists = ll + (size_t)B_ * DIM_;         // B*K floats

  int nInit = B_ * DIM_;
  kl_init_kernel<<<(nInit + 255) / 256, 256, 0, stream>>>(logits, ll, dists);

  const int tiles  = (B_ * K_) / 16;                      // 128
  const int blocks = tiles * (DCHUNKS / WAVES_PER_BLOCK); // 512
  kl_wmma_kernel<<<blocks, 256, 0, stream>>>(qa, ll, dists);

  kl_topk_kernel<<<1, 32, 0, stream>>>(dists, labels, out);
}